// LSTMAutoencoder_57217554317558
// MI455X (gfx1250) — compile-verified
//
#include <hip/hip_runtime.h>
#include <math.h>

#define T_SEQ 4096
#define DIM_I 1024
#define DIM_H 1024
#define G3H   3072

typedef float v2f __attribute__((ext_vector_type(2)));
typedef float v8f __attribute__((ext_vector_type(8)));

__device__ __forceinline__ float sigm(float x) { return 1.0f / (1.0f + expf(-x)); }

__device__ __forceinline__ v8f wmma4(v2f a, v2f b, v8f c) {
  // V_WMMA_F32_16X16X4_F32 : D(16x16,f32) = A(16x4,f32) x B(4x16,f32) + C
  return __builtin_amdgcn_wmma_f32_16x16x4_f32(false, a, false, b, (short)0, c, false, false);
}

// ---- device-wide sense-relative barrier (persistent kernels) ----------------
__device__ __forceinline__ void gsync(unsigned* cnt, unsigned* gen, unsigned nb) {
  __syncthreads();
  if (threadIdx.x == 0) {
    __threadfence();
    unsigned g = __hip_atomic_load(gen, __ATOMIC_RELAXED, __HIP_MEMORY_SCOPE_AGENT);
    unsigned a = __hip_atomic_fetch_add(cnt, 1u, __ATOMIC_ACQ_REL, __HIP_MEMORY_SCOPE_AGENT);
    if (a == nb - 1u) {
      __hip_atomic_store(cnt, 0u, __ATOMIC_RELAXED, __HIP_MEMORY_SCOPE_AGENT);
      __hip_atomic_fetch_add(gen, 1u, __ATOMIC_ACQ_REL, __HIP_MEMORY_SCOPE_AGENT);
    } else {
      while (__hip_atomic_load(gen, __ATOMIC_ACQUIRE, __HIP_MEMORY_SCOPE_AGENT) == g)
        __builtin_amdgcn_s_sleep(2);
    }
  }
  __syncthreads();
}

// ---- big parallel GEMM via fp32 WMMA: C[M,N] = A[M,K] @ op(B) + bias --------
// TRANSB=true : B is row-major [N,K] (weight layout, computes A @ B^T)
// TRANSB=false: B is row-major [K,N] (computes A @ B)
template <bool TRANSB>
__global__ void gemm16_wmma(const float* __restrict__ A, const float* __restrict__ B,
                            const float* __restrict__ bias, float* __restrict__ C,
                            int M, int N, int K) {
  const int lane  = threadIdx.x & 31;
  const int wid   = (int)((blockIdx.x * blockDim.x + threadIdx.x) >> 5);
  const int ntn   = N >> 4;
  const int tiles = (M >> 4) * ntn;
  if (wid >= tiles) return;
  const int m0 = (wid / ntn) << 4;
  const int n0 = (wid % ntn) << 4;
  const int l15   = lane & 15;
  const int lhalf = lane >> 4;       // 0: K={0,1}, 1: K={2,3}
  const int kh    = lhalf << 1;

  const float* arow = A + (size_t)(m0 + l15) * K + kh;
  v8f acc0 = {}, acc1 = {};
  for (int k = 0; k < K; k += 8) {
    v2f a0 = *(const v2f*)(arow + k);
    v2f a1 = *(const v2f*)(arow + k + 4);
    v2f b0, b1;
    if (TRANSB) {
      const float* brow = B + (size_t)(n0 + l15) * K + kh;
      b0 = *(const v2f*)(brow + k);
      b1 = *(const v2f*)(brow + k + 4);
    } else {
      const float* bcol = B + n0 + l15;
      b0.x = bcol[(size_t)(k + kh) * N];
      b0.y = bcol[(size_t)(k + kh + 1) * N];
      b1.x = bcol[(size_t)(k + 4 + kh) * N];
      b1.y = bcol[(size_t)(k + 4 + kh + 1) * N];
    }
    acc0 = wmma4(a0, b0, acc0);
    acc1 = wmma4(a1, b1, acc1);
  }
  v8f acc = acc0 + acc1;
  const float bv = bias ? bias[n0 + l15] : 0.0f;
#pragma unroll
  for (int v = 0; v < 8; ++v) {
    const int row = m0 + v + (lhalf << 3);     // lanes 0-15: M=v ; lanes 16-31: M=v+8
    C[(size_t)row * N + n0 + l15] = acc[v] + bv;
  }
}

// ---- serial matvec tile via WMMA: returns y[n0+lane] for lanes 0..15 --------
// A carries x only in row M=0 (lanes 0 and 16). Loads are unconditional
// (in-bounds for every lane) and zeroed with a branch-free lane select so the
// inner loop has no EXEC-mask manipulation.
__device__ __forceinline__ float matvec_tile(const float* __restrict__ W,
                                             const float* __restrict__ x,
                                             int n0, int K, int lane) {
  const int l15   = lane & 15;
  const int kh    = (lane >> 4) << 1;
  const bool carry = (l15 == 0);               // lanes 0 and 16 carry row M=0
  const float* wrow = W + (size_t)(n0 + l15) * K + kh;
  const float* xp   = x + kh;
  const v2f zero = {0.0f, 0.0f};
  v8f acc0 = {}, acc1 = {};
#pragma unroll 4
  for (int k = 0; k < K; k += 8) {
    v2f xa = *(const v2f*)(xp + k);
    v2f xb = *(const v2f*)(xp + k + 4);
    v2f a0 = carry ? xa : zero;                // v_cndmask, no exec juggling
    v2f a1 = carry ? xb : zero;
    v2f b0 = *(const v2f*)(wrow + k);
    v2f b1 = *(const v2f*)(wrow + k + 4);
    acc0 = wmma4(a0, b0, acc0);
    acc1 = wmma4(a1, b1, acc1);
  }
  return acc0[0] + acc1[0];                    // D(M=0, N=lane) for lanes 0-15
}

// ---- persistent encoder: block = 16 rows x 3 gate-waves, 1 barrier/step -----
__global__ void gru_encoder(const float* __restrict__ W_hh, const float* __restrict__ b_hh,
                            const float* __restrict__ GI,    // [T,3H] incl. b_ih
                            float* __restrict__ h0, float* __restrict__ h1,
                            float* __restrict__ task_emb,    // [T,H] (d_out)
                            unsigned* cnt, unsigned* gen) {
  __shared__ float sgh[48];                    // 3 gates x 16 rows
  const int tid   = (int)threadIdx.x;          // 0..95
  const int lane  = tid & 31;
  const int w     = tid >> 5;                  // gate 0..2
  const int jbase = (int)blockIdx.x << 4;      // 64 blocks x 16 rows = 1024
  const int n0    = w * DIM_H + jbase;         // rows of W_hh for this gate
  const unsigned nb = gridDim.x;

  if (tid < 16) h0[jbase + tid] = 0.0f;        // h_{-1} = 0 (buffer 0)
  gsync(cnt, gen, nb);

  for (int t = 0; t < T_SEQ; ++t) {
    const float* hin  = (t & 1) ? h1 : h0;     // double-buffered hidden state
    float*       hout = (t & 1) ? h0 : h1;
    float v = matvec_tile(W_hh, hin, n0, DIM_H, lane) + b_hh[n0 + (lane & 15)];
    if (lane < 16) sgh[(w << 4) + lane] = v;
    __syncthreads();
    if (tid < 16) {
      const int j = jbase + tid;
      const float* gi = GI + (size_t)t * G3H;
      const float r  = sigm(gi[j] + sgh[tid]);
      const float z  = sigm(gi[j + DIM_H] + sgh[16 + tid]);
      const float n  = tanhf(gi[j + 2 * DIM_H] + r * sgh[32 + tid]);
      const float hn = (1.0f - z) * n + z * hin[j];
      hout[j] = hn;
      task_emb[(size_t)t * DIM_H + j] = hn;
    }
    gsync(cnt, gen, nb);                       // single device barrier per step
  }
}
// T even -> final hidden state ends in h0, where the decoder starts.

// ---- persistent decoder: block = 16 rows x 6 gate-waves, 1 barrier/step -----
__global__ void gru_decoder(const float* __restrict__ W_hh, const float* __restrict__ b_hh,
                            const float* __restrict__ Wc,   const float* __restrict__ bc,
                            const float* __restrict__ b_ih0,  // gi at t==0 (x=0)
                            float* __restrict__ h0, float* __restrict__ h1,
                            float* __restrict__ hdec,         // [T,H]
                            unsigned* cnt, unsigned* gen) {
  __shared__ float sg[96];                     // [0:48)=gh gates, [48:96)=gi gates
  const int tid   = (int)threadIdx.x;          // 0..191
  const int lane  = tid & 31;
  const int w     = tid >> 5;                  // 0..5
  const int jbase = (int)blockIdx.x << 4;
  const unsigned nb = gridDim.x;

  for (int t = 0; t < T_SEQ; ++t) {
    const float* hin  = (t & 1) ? h1 : h0;
    float*       hout = (t & 1) ? h0 : h1;
    if (w < 3) {                               // gh = W_hh h + b_hh
      const int n0 = w * DIM_H + jbase;
      float v = matvec_tile(W_hh, hin, n0, DIM_H, lane) + b_hh[n0 + (lane & 15)];
      if (lane < 16) sg[(w << 4) + lane] = v;
    } else {                                   // gi = Wc h + bc  (x folded thru fc)
      const int g  = w - 3;
      const int n0 = g * DIM_H + jbase;
      float v;
      if (t == 0) v = b_ih0[n0 + (lane & 15)]; // x_0 = 0 -> gi = b_ih
      else        v = matvec_tile(Wc, hin, n0, DIM_H, lane) + bc[n0 + (lane & 15)];
      if (lane < 16) sg[48 + (g << 4) + lane] = v;
    }
    __syncthreads();
    if (tid < 16) {
      const int j = jbase + tid;
      const float r  = sigm(sg[48 + tid] + sg[tid]);
      const float z  = sigm(sg[64 + tid] + sg[16 + tid]);
      const float n  = tanhf(sg[80 + tid] + r * sg[32 + tid]);
      const float hn = (1.0f - z) * n + z * hin[j];
      hout[j] = hn;
      hdec[(size_t)t * DIM_H + j] = hn;
    }
    gsync(cnt, gen, nb);
  }
}

// ---- small helpers ----------------------------------------------------------
__global__ void init_bar(unsigned* cnt) { if (threadIdx.x == 0) *cnt = 0u; }

__global__ void bc_kernel(const float* __restrict__ dec_w_ih, const float* __restrict__ fc_b,
                          const float* __restrict__ dec_b_ih, float* __restrict__ bc) {
  const int n = (int)(blockIdx.x * blockDim.x + threadIdx.x);
  if (n >= G3H) return;
  const float* w = dec_w_ih + (size_t)n * DIM_I;
  float s = dec_b_ih[n];
  for (int j = 0; j < DIM_I; ++j) s += w[j] * fc_b[j];
  bc[n] = s;
}

__global__ void loss_rows(const float* __restrict__ inputs, const float* __restrict__ decx,
                          float* __restrict__ partials) {
  const int t = (int)blockIdx.x;
  const float* a = inputs + (size_t)(T_SEQ - 1 - t) * DIM_I;  // reversed compare
  const float* b = decx + (size_t)t * DIM_I;
  float s = 0.0f;
  for (int j = threadIdx.x; j < DIM_I; j += blockDim.x) {
    const float d = a[j] - b[j];
    s += d * d;
  }
  __shared__ float red[256];
  red[threadIdx.x] = s; __syncthreads();
  for (int st = 128; st > 0; st >>= 1) {
    if ((int)threadIdx.x < st) red[threadIdx.x] += red[threadIdx.x + st];
    __syncthreads();
  }
  if (threadIdx.x == 0) partials[t] = red[0];
}

__global__ void finalize_loss(const float* __restrict__ partials, float* __restrict__ loss) {
  float s = 0.0f;
  for (int t = threadIdx.x; t < T_SEQ; t += blockDim.x) s += partials[t];
  __shared__ float red[256];
  red[threadIdx.x] = s; __syncthreads();
  for (int st = 128; st > 0; st >>= 1) {
    if ((int)threadIdx.x < st) red[threadIdx.x] += red[threadIdx.x + st];
    __syncthreads();
  }
  if (threadIdx.x == 0) *loss = red[0] / (float)((size_t)T_SEQ * DIM_I);
}

__global__ void col_mean(const float* __restrict__ emb, float* __restrict__ out) {
  const int c = (int)(blockIdx.x * blockDim.x + threadIdx.x);
  if (c >= DIM_H) return;
  float s = 0.0f;
  for (int t = 0; t < T_SEQ; ++t) s += emb[(size_t)t * DIM_H + c];
  out[c] = s / (float)T_SEQ;
}

// ---- launcher ---------------------------------------------------------------
extern "C" void kernel_launch(void* const* d_in, const int* in_sizes, int n_in,
                              void* d_out, int out_size, void* d_ws, size_t ws_size,
                              hipStream_t stream) {
  const float* inputs   = (const float*)d_in[0];
  const float* enc_w_ih = (const float*)d_in[1];
  const float* enc_w_hh = (const float*)d_in[2];
  const float* enc_b_ih = (const float*)d_in[3];
  const float* enc_b_hh = (const float*)d_in[4];
  const float* dec_w_ih = (const float*)d_in[5];
  const float* dec_w_hh = (const float*)d_in[6];
  const float* dec_b_ih = (const float*)d_in[7];
  const float* dec_b_hh = (const float*)d_in[8];
  const float* fc_w     = (const float*)d_in[9];
  const float* fc_b     = (const float*)d_in[10];

  float* out       = (float*)d_out;
  float* emb_vec   = out;                                    // [H]
  float* task_emb  = out + DIM_H;                            // [T,H]
  float* loss      = out + DIM_H + (size_t)T_SEQ * DIM_H;    // [1]

  float* ws = (float*)d_ws;
  size_t off = 0;
  float* gi_enc = ws + off; off += (size_t)T_SEQ * G3H;      // encoder gi, all steps
  float* Wc     = ws + off; off += (size_t)G3H * DIM_H;      // dec_w_ih @ fc_w
  float* bc     = ws + off; off += G3H;
  float* h0     = ws + off; off += DIM_H;                    // double-buffered h
  float* h1     = ws + off; off += DIM_H;
  float* hdec   = ws + off; off += (size_t)T_SEQ * DIM_H;
  float* decx   = ws + off; off += (size_t)T_SEQ * DIM_H;
  float* part   = ws + off; off += T_SEQ;
  unsigned* cnt = (unsigned*)(ws + off);
  unsigned* gen = cnt + 1;

  init_bar<<<1, 64, 0, stream>>>(cnt);

  // Phase 1 (parallel): GI_enc = inputs @ enc_w_ih^T + b_ih    [4096 x 3072]
  gemm16_wmma<true><<<6144, 256, 0, stream>>>(inputs, enc_w_ih, enc_b_ih, gi_enc,
                                              T_SEQ, G3H, DIM_I);
  // Fold decoder feedback: Wc = dec_w_ih @ fc_w  [3072 x 1024]; bc = b_ih + W_ih@fc_b
  gemm16_wmma<false><<<1536, 256, 0, stream>>>(dec_w_ih, fc_w, nullptr, Wc,
                                               G3H, DIM_H, DIM_I);
  bc_kernel<<<12, 256, 0, stream>>>(dec_w_ih, fc_b, dec_b_ih, bc);

  // Phase 2 (serial): encoder recurrence. 64 blocks x 3 gate-waves,
  // gates staged in LDS, one device barrier per step.
  gru_encoder<<<64, 96, 0, stream>>>(enc_w_hh, enc_b_hh, gi_enc, h0, h1,
                                     task_emb, cnt, gen);

  // Phase 3 (serial): decoder recurrence. 64 blocks x 6 gate-waves.
  gru_decoder<<<64, 192, 0, stream>>>(dec_w_hh, dec_b_hh, Wc, bc, dec_b_ih,
                                      h0, h1, hdec, cnt, gen);

  // Phase 4 (parallel): dec_seq = hdec @ fc_w^T + fc_b        [4096 x 1024]
  gemm16_wmma<true><<<2048, 256, 0, stream>>>(hdec, fc_w, fc_b, decx,
                                              T_SEQ, DIM_H, DIM_H);

  // Outputs: reversed-MSE loss + per-channel mean of task_emb (deterministic trees)
  loss_rows<<<T_SEQ, 256, 0, stream>>>(inputs, decx, part);
  finalize_loss<<<1, 256, 0, stream>>>(part, loss);
  col_mean<<<4, 256, 0, stream>>>(task_emb, emb_vec);
}